// NCC_vxm_33655363732036
// MI455X (gfx1250) — compile-verified
//
#include <hip/hip_runtime.h>

// ---- problem constants (shape (2,1,160,192,160), 9^3 box, zero "same" pad) ----
#define WIN   9
#define HALO  4
#define TILE  16
#define IN    24            // TILE + 2*HALO
#define ND    160
#define NH    192
#define NW    160
#define NBATCH 2

#define SZ_IN (IN*IN*IN)          // 13824 floats  (54.0 KB)
#define SZ_T1 (IN*IN*TILE)        //  9216 floats  (36.9 KB)
#define SZ_T2 (IN*TILE*TILE)      //  6144 floats  (24.6 KB)
#define SZ_S  (TILE*TILE*TILE)    //  4096 floats  (16.4 KB)
#define LDS_FLOATS (2*SZ_IN + SZ_T1 + SZ_T2 + 5*SZ_S)   // 63488 floats = 248 KB < 320 KB/WGP

typedef float v2f __attribute__((ext_vector_type(2)));
typedef float v8f __attribute__((ext_vector_type(8)));
typedef int   v4i __attribute__((ext_vector_type(4)));

// ---- gfx1250 async global->LDS staging (GLOBAL_LOAD_ASYNC_TO_LDS_B128) ----
// Parameter types per hipcc diagnostic: (AS1 v4i*, AS3 v4i*, imm offset, imm cpol).
#if defined(__AMDGCN__) && __has_builtin(__builtin_amdgcn_global_load_async_to_lds_b128)
#define G2L_ASYNC4(gp, lp)                                                  \
  __builtin_amdgcn_global_load_async_to_lds_b128(                           \
      (__attribute__((address_space(1))) v4i*)(gp),                         \
      (__attribute__((address_space(3))) v4i*)(lp), 0, 0)
#define HAVE_ASYNC_LDS 1
#else
#define G2L_ASYNC4(gp, lp) (*(float4*)(lp) = *(const float4*)(gp))
#define HAVE_ASYNC_LDS 0
#endif

#if HAVE_ASYNC_LDS && __has_builtin(__builtin_amdgcn_s_wait_asynccnt)
#define WAIT_ASYNC() __builtin_amdgcn_s_wait_asynccnt(0)
#elif HAVE_ASYNC_LDS
#define WAIT_ASYNC() asm volatile("s_wait_asynccnt 0" ::: "memory")
#else
#define WAIT_ASYNC() ((void)0)
#endif

// Fused NCC tile kernel: one workgroup produces one 16^3 output tile.
// Separable 9-wide box sums are banded ones-matrix multiplies done with
// V_WMMA_F32_16X16X4_F32 (6 K=4 chunks cover the 24-wide haloed input).
__global__ __launch_bounds__(256) void ncc_tile_kernel(
    const float* __restrict__ yt, const float* __restrict__ yp,
    float* __restrict__ partials)
{
  extern __shared__ float smem[];
  float* sI = smem;                 // 24^3 I tile
  float* sJ = sI + SZ_IN;           // 24^3 J tile
  float* T1 = sJ + SZ_IN;           // after W pass: [d24][h24][w16]
  float* T2 = T1 + SZ_T1;           // after H pass: [d24][h16][w16]
  float* S  = T2 + SZ_T2;           // 5 final box-sum fields [16^3]

  const int tid  = threadIdx.x;
  const int lane = tid & 31;        // wave32
  const int wid  = tid >> 5;        // 8 waves / block
  const int half = lane >> 4;       // 0: lanes 0-15, 1: lanes 16-31
  const int mn   = lane & 15;       // row (A) / col (B,C,D) index

  const int nb = (int)blockIdx.z / (ND / TILE);
  const int dt = (int)blockIdx.z % (ND / TILE);
  const int d0 = dt * TILE - HALO;
  const int h0 = (int)blockIdx.y * TILE - HALO;
  const int w0 = (int)blockIdx.x * TILE - HALO;

  // ---- stage I and J (24^3 with zero halo) into LDS, 16B per lane async ----
  // Each float4 group is entirely in-range or entirely out-of-range in W
  // (groups are 4-aligned and the W tile origin is 4-aligned), so per-group
  // predication implements the zero "same" padding exactly.
  for (int v = tid; v < SZ_IN / 4; v += 256) {       // 3456 float4 groups
    int row = v / 6;                                 // flat (d,h): 0..575
    int g   = v - row * 6;                           // group in row: 0..5
    int ld  = row / IN;
    int lh  = row - ld * IN;
    int lw  = 4 * g;
    int gd = d0 + ld, gh = h0 + lh, gw = w0 + lw;
    bool inr = ((unsigned)gd < ND) && ((unsigned)gh < NH) && ((unsigned)gw < NW);
    int li = row * IN + lw;                          // LDS offset (16B aligned)
    if (inr) {
      size_t idx = (((size_t)nb * ND + gd) * NH + gh) * NW + gw;  // 16B aligned
      G2L_ASYNC4(yt + idx, sI + li);
      G2L_ASYNC4(yp + idx, sJ + li);
    } else {
      float4 z = {0.f, 0.f, 0.f, 0.f};
      *(float4*)(sI + li) = z;
      *(float4*)(sJ + li) = z;
    }
  }

  // ---- banded ones matrix, six 16x4 chunks; identical per-lane values serve
  //      as A-operand (Band, H/D passes) and B-operand (Band^T, W pass) ----
  v2f bnd[6];
#pragma unroll
  for (int c = 0; c < 6; ++c) {
    int k0 = 4 * c + 2 * half;
    bnd[c].x = (k0     >= mn && k0     <= mn + 8) ? 1.f : 0.f;
    bnd[c].y = (k0 + 1 >= mn && k0 + 1 <= mn + 8) ? 1.f : 0.f;
  }

  WAIT_ASYNC();
  __syncthreads();

#pragma unroll   // force compile-time field selection (no runtime branches)
  for (int f = 0; f < 5; ++f) {     // I, J, I*I, J*J, I*J
    // ---- W pass: O(16x16) = Data(16x24) x Band^T(24x16), rows = flat (d,h) ----
    for (int t = wid; t < (IN * IN) / 16; t += 8) {  // 576/16 = 36 row-tiles
      int rowm = t * 16 + mn;
      int dd = rowm / IN;
      int hh = rowm - dd * IN;
      const float* rowI = &sI[(dd * IN + hh) * IN];
      const float* rowJ = &sJ[(dd * IN + hh) * IN];
      v8f acc = {};
#pragma unroll
      for (int c = 0; c < 6; ++c) {
        int col = 4 * c + 2 * half;                  // even -> 8B aligned
        float2 iv = *(const float2*)&rowI[col];
        float2 jv = *(const float2*)&rowJ[col];
        v2f a;
        if (f == 0)      { a.x = iv.x;        a.y = iv.y; }
        else if (f == 1) { a.x = jv.x;        a.y = jv.y; }
        else if (f == 2) { a.x = iv.x * iv.x; a.y = iv.y * iv.y; }
        else if (f == 3) { a.x = jv.x * jv.x; a.y = jv.y * jv.y; }
        else             { a.x = iv.x * jv.x; a.y = iv.y * jv.y; }
        acc = __builtin_amdgcn_wmma_f32_16x16x4_f32(
            false, a, false, bnd[c], (short)0, acc, false, false);
      }
#pragma unroll
      for (int r = 0; r < 8; ++r) {
        int flat = t * 16 + r + 8 * half;            // D layout: M = r + 8*half
        int dd2 = flat / IN;
        int hh2 = flat - dd2 * IN;
        T1[(dd2 * IN + hh2) * TILE + mn] = acc[r];
      }
    }
    __syncthreads();

    // ---- H pass: O(16x16) = Band(16x24) x T1[d][:][:](24x16), per d ----
    for (int dd = wid; dd < IN; dd += 8) {           // 24 matrices
      v8f acc = {};
#pragma unroll
      for (int c = 0; c < 6; ++c) {
        int k0 = 4 * c + 2 * half;
        v2f b;
        b.x = T1[(dd * IN + k0    ) * TILE + mn];
        b.y = T1[(dd * IN + k0 + 1) * TILE + mn];
        acc = __builtin_amdgcn_wmma_f32_16x16x4_f32(
            false, bnd[c], false, b, (short)0, acc, false, false);
      }
#pragma unroll
      for (int r = 0; r < 8; ++r) {
        int ho = r + 8 * half;
        T2[(dd * TILE + ho) * TILE + mn] = acc[r];
      }
    }
    __syncthreads();

    // ---- D pass: O(16x16) = Band(16x24) x T2[:][h][:](24x16), per h ----
    for (int hh = wid; hh < TILE; hh += 8) {         // 16 matrices
      v8f acc = {};
#pragma unroll
      for (int c = 0; c < 6; ++c) {
        int k0 = 4 * c + 2 * half;
        v2f b;
        b.x = T2[( k0      * TILE + hh) * TILE + mn];
        b.y = T2[((k0 + 1) * TILE + hh) * TILE + mn];
        acc = __builtin_amdgcn_wmma_f32_16x16x4_f32(
            false, bnd[c], false, b, (short)0, acc, false, false);
      }
#pragma unroll
      for (int r = 0; r < 8; ++r) {
        int dout = r + 8 * half;
        S[f * SZ_S + (dout * TILE + hh) * TILE + mn] = acc[r];
      }
    }
    __syncthreads();
  }

  // ---- pointwise combine (matches reference math) + block reduction ----
  float local = 0.f;
  const float inv_win = 1.0f / 729.0f;               // 1 / 9^3
  for (int v = tid; v < SZ_S; v += 256) {
    float Is = S[v];
    float Js = S[SZ_S + v];
    float I2 = S[2 * SZ_S + v];
    float J2 = S[3 * SZ_S + v];
    float IJ = S[4 * SZ_S + v];
    float uI = Is * inv_win;
    float uJ = Js * inv_win;
    float cross = IJ - uI * Js;
    float Iv = fmaxf(I2 - uI * Is, 1e-5f);
    float Jv = fmaxf(J2 - uJ * Js, 1e-5f);
    local += (cross * cross) / (Iv * Jv + 1e-5f);
  }
  float* red = smem;                                 // reuse sI region (S untouched)
  red[tid] = local;
  __syncthreads();
  for (int s = 128; s > 0; s >>= 1) {
    if (tid < s) red[tid] += red[tid + s];
    __syncthreads();
  }
  if (tid == 0) {
    int wg = (int)blockIdx.x +
             (int)gridDim.x * ((int)blockIdx.y + (int)gridDim.y * (int)blockIdx.z);
    partials[wg] = red[0];
  }
}

// Deterministic final reduction of per-workgroup partials -> -sum(cc).
__global__ __launch_bounds__(256) void ncc_reduce_kernel(
    const float* __restrict__ partials, int n, float* __restrict__ out)
{
  __shared__ float red[256];
  float s = 0.f;
  for (int i = threadIdx.x; i < n; i += 256) s += partials[i];
  red[threadIdx.x] = s;
  __syncthreads();
  for (int k = 128; k > 0; k >>= 1) {
    if ((int)threadIdx.x < k) red[threadIdx.x] += red[threadIdx.x + k];
    __syncthreads();
  }
  if (threadIdx.x == 0) out[0] = -red[0];
}

extern "C" void kernel_launch(void* const* d_in, const int* in_sizes, int n_in,
                              void* d_out, int out_size, void* d_ws, size_t ws_size,
                              hipStream_t stream) {
  (void)in_sizes; (void)n_in; (void)out_size; (void)ws_size;
  const float* yt = (const float*)d_in[0];   // y_true, f32 (2,1,160,192,160)
  const float* yp = (const float*)d_in[1];   // y_pred, f32
  float* out = (float*)d_out;                // single f32 scalar
  float* partials = (float*)d_ws;            // 2400 floats of scratch

  dim3 grid(NW / TILE, NH / TILE, NBATCH * (ND / TILE));   // (10, 12, 20)
  const int nblocks = (NW / TILE) * (NH / TILE) * NBATCH * (ND / TILE); // 2400
  size_t shmem = (size_t)LDS_FLOATS * sizeof(float);       // 253,952 B

  ncc_tile_kernel<<<grid, 256, shmem, stream>>>(yt, yp, partials);
  ncc_reduce_kernel<<<1, 256, 0, stream>>>(partials, nblocks, out);
}